// RoiPooling_49340584296964
// MI455X (gfx1250) — compile-verified
//
#include <hip/hip_runtime.h>
#include <cstdint>
#include <math.h>

// RoiPooling (adaptive 7x7 max pool over quantized boxes)
// feature: [1, 256, 50, 50] f32   proposal: [64, 4] f32
// out:     [64, 256, 7, 7]  f32
//
// One workgroup per ROI, 256 threads (thread == channel).
// The entire 8x256x8 ROI tile is staged into LDS with a single CDNA5
// Tensor Data Mover instruction (tensor_load_to_lds, TENSORcnt), then each
// thread computes its channel's 49 adaptive bin maxes from LDS.

#define ROI_N 64
#define CCH   256
#define FH    50
#define FW    50
#define PS    7

typedef uint32_t u32x4 __attribute__((ext_vector_type(4)));
typedef uint32_t u32x8 __attribute__((ext_vector_type(8)));

__global__ __launch_bounds__(256) void roi_maxpool_gfx1250(
    const float* __restrict__ feature,   // [256,50,50]
    const float* __restrict__ proposal,  // [64,4]
    float* __restrict__ out)             // [64,256,7,7]
{
    // TDM writes the tile packed in loop order x(w=8) inner, y(c=256), z(h=8):
    // element (w,c,h) lands at dword (h*256 + c)*8 + w. Lane stride = 8 dwords
    // -> 4-way LDS bank conflicts at worst in the compute phase, 64 KB total.
    __shared__ float smem[8 * CCH * 8];

    const int roi = blockIdx.x;
    const int c   = threadIdx.x;         // 256 threads == 256 channels

    // ---- block-uniform ROI parameters (scalar FP pipeline) ----
    const float scale = 1.0f / 16.0f;
    int x1 = (int)floorf(proposal[roi * 4 + 0] * scale);
    int y1 = (int)floorf(proposal[roi * 4 + 1] * scale);
    int x2 = (int)floorf(proposal[roi * 4 + 2] * scale);
    int y2 = (int)floorf(proposal[roi * 4 + 3] * scale);
    int lh = y2 - y1 + 1;
    int lw = x2 - x1 + 1;
    // Defensive clamps (reference inputs guarantee 1..8, in-bounds).
    lh = max(1, min(lh, 8));
    lw = max(1, min(lw, 8));
    x1 = max(0, min(x1, FW - 1));
    y1 = max(0, min(y1, FH - 1));
    lw = min(lw, FW - x1);
    lh = min(lh, FH - y1);

    // LDS byte offset of smem: flat->LDS translation uses addr[31:0]; the
    // shared aperture's low 32 bits are zero, so truncation yields the DS addr.
    const unsigned lds_base = (unsigned)(uintptr_t)(&smem[0]);

    // ---- stage tile with one Tensor Data Mover DMA (wave 0 only) ----
    if (threadIdx.x < 32) {
        const uint64_t gaddr = (uint64_t)(uintptr_t)feature +
                               4ull * (uint64_t)(y1 * FW + x1);   // tile start
        // D# group 0: count=1 | lds_addr | global_addr(57b) | type=2
        const u32x4 g0 = {
            1u,                                            // count=1 (valid)
            lds_base,                                      // lds_addr (bytes)
            (uint32_t)gaddr,                               // global_addr[31:0]
            (uint32_t)((gaddr >> 32) & 0x01FFFFFFu) | (2u << 30) // [120:64]|type
        };
        // D# group 1: data_size=4B; dims measured from tile start (OOB -> 0,
        // written only to never-read slots). x=w, y=c, z=h.
        const uint32_t tdim0 = (uint32_t)(FW - x1);        // x extent (w)
        const uint32_t tdim1 = CCH;                        // y extent (c)
        const u32x8 g1 = {
            2u << 16,                                      // data_size=2 (4B)
            (tdim0 & 0xFFFFu) << 16,                       // abar=0 | tdim0.lo
            (tdim0 >> 16) | ((tdim1 & 0xFFFFu) << 16),     // tdim0.hi | tdim1.lo
            (tdim1 >> 16) | (8u << 16),                    // tdim1.hi | tile_dim0=8
            256u | (8u << 16),                             // tile_dim1=256 | tile_dim2=8
            2500u,                                         // d0 stride (c step) lo32
            (50u << 16),                                   // d0s.hi=0 | d1 stride (h step).lo16
            0u                                             // d1 stride hi
        };
        // D# group 2: tensor_dim2 (z extent), no dim3, iterate off.
        const u32x4 g2 = { (uint32_t)(FH - y1), 0u, 0u, 0u };
        // D# group 3: unused higher dims.
        const u32x4 g3 = { 0u, 0u, 0u, 0u };

        asm volatile("tensor_load_to_lds %0, %1, %2, %3"
                     :: "s"(g0), "s"(g1), "s"(g2), "s"(g3)
                     : "memory");
        __builtin_amdgcn_s_wait_tensorcnt(0);              // data landed in LDS
    }
    __syncthreads();                                        // publish to all 8 waves

    // ---- adaptive 7x7 max pool from LDS ----
    const size_t obase = ((size_t)roi * CCH + (size_t)c) * (PS * PS);
    for (int i = 0; i < PS; ++i) {
        const int hlo = (i * lh) / PS;
        const int hhi = ((i + 1) * lh + (PS - 1)) / PS;     // ceil((i+1)*lh/7)
        for (int j = 0; j < PS; ++j) {
            const int wlo = (j * lw) / PS;
            const int whi = ((j + 1) * lw + (PS - 1)) / PS;
            float m = -INFINITY;                            // empty bin -> -inf
            for (int h = hlo; h < hhi; ++h)
                for (int w = wlo; w < whi; ++w)
                    m = fmaxf(m, smem[(h * CCH + c) * 8 + w]);
            out[obase + (size_t)(i * PS + j)] = m;
        }
    }
}

extern "C" void kernel_launch(void* const* d_in, const int* in_sizes, int n_in,
                              void* d_out, int out_size, void* d_ws, size_t ws_size,
                              hipStream_t stream) {
    (void)in_sizes; (void)n_in; (void)out_size; (void)d_ws; (void)ws_size;
    const float* feature  = (const float*)d_in[0];
    const float* proposal = (const float*)d_in[1];
    float*       out      = (float*)d_out;
    roi_maxpool_gfx1250<<<dim3(ROI_N), dim3(CCH), 0, stream>>>(feature, proposal, out);
}